// BoundaryAwareLoss_81527069213368
// MI455X (gfx1250) — compile-verified
//
#include <hip/hip_runtime.h>

// Problem constants (reference: N=16, C=2, H=W=1024)
#define Hh 1024
#define Ww 1024
#define TILE 64
#define HALO_H 68    // TILE + 2*2 (y halo, origin -2)
#define ROW_U 18     // uints per LDS tile row: 72 cols, x origin -4 (16B aligned)

typedef float v2f __attribute__((ext_vector_type(2)));
typedef float v8f __attribute__((ext_vector_type(8)));
typedef float f4  __attribute__((ext_vector_type(4)));
typedef int   i4  __attribute__((ext_vector_type(4)));

#if defined(__HIP_DEVICE_COMPILE__) && defined(__has_builtin)
#if __has_builtin(__builtin_amdgcn_wmma_f32_16x16x4_f32)
#define USE_WMMA 1
#endif
#if __has_builtin(__builtin_amdgcn_exp2f) && __has_builtin(__builtin_amdgcn_logf)
#define FAST_TRANS 1
#endif
#endif
#ifndef USE_WMMA
#define USE_WMMA 0
#endif

__device__ __forceinline__ float fast_exp2(float x) {
#if defined(FAST_TRANS)
    return __builtin_amdgcn_exp2f(x);   // raw v_exp_f32
#else
    return exp2f(x);
#endif
}
__device__ __forceinline__ float fast_log2(float x) {
#if defined(FAST_TRANS)
    return __builtin_amdgcn_logf(x);    // raw v_log_f32
#else
    return log2f(x);
#endif
}

extern "C" __global__ __launch_bounds__(256)
void boundary_ce_main(const float* __restrict__ logits,
                      const int*   __restrict__ tgt,
                      float*       __restrict__ blockPartials)
{
    // Flag bytes, 4 per uint. Per byte:
    //  bit0 = road & inb          (dilate input; OOB -> 0 = OR identity)
    //  bit1 = road | !inb         (erode  input; OOB -> 1 = AND identity)
    //  bit2 = valid (label != 255)
    //  bit3 = target channel (clip(label,0,1))
    __shared__ unsigned int tileW[HALO_H * ROW_U];
    __shared__ unsigned int rowp[HALO_H * (TILE / 4)]; // horiz 5-tap: or | and<<1
    __shared__ float wavePart[16];

    const int tid = threadIdx.x;
    const int tx = blockIdx.x, ty = blockIdx.y, n = blockIdx.z;
    const int gx0 = tx * TILE - 4;   // aligned tile x origin
    const int gy0 = ty * TILE - 2;
    const long tbase = (long)n * (Hh * Ww);

    // ---- Phase 1: aligned b128 target loads + SWAR flag packing ----
    for (int q = tid; q < HALO_H * ROW_U; q += 256) {
        int hy = q / ROW_U;
        int k  = q - hy * ROW_U;           // quad index within row (0..17)
        int gy = gy0 + hy;
        int cy = min(max(gy, 0), Hh - 1);
        int qs = gx0 + 4 * k;              // quad start col (multiple of 4)
        int cqs = min(max(qs, 0), Ww - 4);
        // x-OOB only occurs in whole quads (origin & width are 4-aligned)
        unsigned int inb01 = ((unsigned)gy < (unsigned)Hh &&
                              (unsigned)qs <= (unsigned)(Ww - 4)) ? 0x01010101u : 0u;
        i4 lv = *(const i4*)(tgt + tbase + (long)cy * Ww + cqs);
        // pack label LSB bytes; labels are in {0,1,255}
        unsigned int p8 = ((unsigned)lv.x & 0xFFu)
                        | (((unsigned)lv.y & 0xFFu) << 8)
                        | (((unsigned)lv.z & 0xFFu) << 16)
                        | (((unsigned)lv.w & 0xFFu) << 24);
        unsigned int t_b = p8 & 0x01010101u;                 // label >= 1
        unsigned int v_b = ~(p8 >> 1) & 0x01010101u;         // label != 255
        unsigned int r_b = t_b & v_b;                        // label == 1
        unsigned int b0  = r_b & inb01;
        unsigned int b1  = r_b | (0x01010101u ^ inb01);
        tileW[q] = b0 | (b1 << 1) | (v_b << 2) | (t_b << 3);
    }
    __syncthreads();

    // ---- Phase 2: horizontal 5-tap OR/AND via 64-bit SWAR, 4 cols/iter ----
    for (int q = tid; q < HALO_H * (TILE / 4); q += 256) {
        int ry  = q >> 4;    // TILE/4 == 16
        int rqx = q & 15;
        // window: tile cols 4*rqx+2 .. 4*rqx+9 (output cols 4rqx-2..4rqx+5)
        unsigned int u0 = tileW[ry * ROW_U + rqx];
        unsigned int u1 = tileW[ry * ROW_U + rqx + 1];
        unsigned int u2 = tileW[ry * ROW_U + rqx + 2];
        unsigned int o0 = u0 & 0x01010101u, a0 = (u0 >> 1) & 0x01010101u;
        unsigned int o1 = u1 & 0x01010101u, a1 = (u1 >> 1) & 0x01010101u;
        unsigned int o2 = u2 & 0x01010101u, a2 = (u2 >> 1) & 0x01010101u;
        unsigned long long O = (unsigned long long)(o0 >> 16)
                             | ((unsigned long long)o1 << 16)
                             | ((unsigned long long)o2 << 48);
        unsigned long long A = (unsigned long long)(a0 >> 16)
                             | ((unsigned long long)a1 << 16)
                             | ((unsigned long long)a2 << 48);
        unsigned long long o5 = O | (O >> 8) | (O >> 16) | (O >> 24) | (O >> 32);
        unsigned long long a5 = A & (A >> 8) & (A >> 16) & (A >> 24) & (A >> 32);
        rowp[q] = (unsigned int)(o5 & 0x01010101ull)
                | ((unsigned int)(a5 & 0x01010101ull) << 1);
    }
    __syncthreads();

    // ---- Phase 3: vertical 5-tap + CE (base-2), WMMA accumulation ----
    const int qx  = tid & 15;   // 4-pixel quad within a row (lane%16 == tid%16)
    const int r0  = tid >> 4;   // base row (0..15)
    const int gxq = tx * TILE + qx * 4;
    const float LOG2E = 1.4426950408889634f;

#if USE_WMMA
    // A selector (16x4 f32 layout): rows m<8 sum B slots K in {0,2} (= b.x ->
    // weighted CE), rows m>=8 sum K in {1,3} (= b.y -> valid count)
    v8f acc = {};
    const float selhi = ((tid & 15) < 8) ? 1.0f : 0.0f;
    v2f asel; asel.x = selhi; asel.y = 1.0f - selhi;
#else
    float wsum = 0.0f, vcnt = 0.0f;
#endif

    #pragma unroll
    for (int rr = 0; rr < 4; ++rr) {
        const int oy = rr * 16 + r0;         // output row within tile (0..63)
        const int gy = ty * TILE + oy;

        unsigned int m_or = 0u, m_and = 0xFFFFFFFFu;
        #pragma unroll
        for (int k = 0; k < 5; ++k) {
            unsigned int r = rowp[(oy + k) * (TILE / 4) + qx];
            m_or  |= r;
            m_and &= r;
        }
        unsigned int bnd4 = (m_or & 0x01010101u) & ~((m_and >> 1) & 0x01010101u);

        // center flag bytes: tile cols 4qx+4..4qx+7 == output cols 4qx..4qx+3
        unsigned int lab4 = tileW[(oy + 2) * ROW_U + qx + 1];

        const f4* p0 = (const f4*)(logits + ((size_t)(n * 2 + 0) * Hh + gy) * Ww + gxq);
        const f4* p1 = (const f4*)(logits + ((size_t)(n * 2 + 1) * Hh + gy) * Ww + gxq);
        f4 x0 = __builtin_nontemporal_load(p0);
        f4 x1 = __builtin_nontemporal_load(p1);

        float ws_it = 0.0f;
        float vc_it = (float)__popc(lab4 & 0x04040404u);
        #pragma unroll
        for (int i = 0; i < 4; ++i) {
            int tsel = (int)((lab4 >> (8 * i + 3)) & 1u);
            int vi   = (int)((lab4 >> (8 * i + 2)) & 1u);
            int bi   = (int)((bnd4 >> (8 * i)) & 1u);
            float t  = (x0[i] - x1[i]) * LOG2E;
            float ts = tsel ? t : -t;                 // (x_other - x_target)*log2e
            // softplus in base 2: log2(1+2^ts) = max(ts,0) + log2(1+2^-|ts|)
            float e  = fast_exp2(-fabsf(ts));
            float sp = fmaxf(ts, 0.0f) + fast_log2(1.0f + e);
            float f  = (float)(vi << bi);             // valid * (1 + boundary)
            ws_it = fmaf(f, sp, ws_it);               // (ln2 applied at the end)
        }

#if USE_WMMA
        v2f bvec; bvec.x = ws_it; bvec.y = vc_it;
        acc = __builtin_amdgcn_wmma_f32_16x16x4_f32(
            false, asel, false, bvec, (short)0, acc, false, false);
#else
        wsum += ws_it; vcnt += vc_it;
#endif
    }

    // ---- block reduction (deterministic; no atomics) ----
    const int lane = tid & 31;
    const int wid  = tid >> 5;
#if USE_WMMA
    // C/D layout: VGPR0 lanes0-15 = row0 (weighted-CE col sums),
    //             lanes16-31 = row8 (valid-count col sums)
    float v = acc[0];
    v += __shfl_xor(v, 1, 32);
    v += __shfl_xor(v, 2, 32);
    v += __shfl_xor(v, 4, 32);
    v += __shfl_xor(v, 8, 32);
    if (lane == 0)  wavePart[wid * 2 + 0] = v;
    if (lane == 16) wavePart[wid * 2 + 1] = v;
#else
    float a = wsum, c = vcnt;
    for (int off = 16; off >= 1; off >>= 1) {
        a += __shfl_xor(a, off, 32);
        c += __shfl_xor(c, off, 32);
    }
    if (lane == 0) { wavePart[wid * 2 + 0] = a; wavePart[wid * 2 + 1] = c; }
#endif
    __syncthreads();
    if (tid == 0) {
        float sw = 0.0f, sv = 0.0f;
        #pragma unroll
        for (int w = 0; w < 8; ++w) { sw += wavePart[w * 2]; sv += wavePart[w * 2 + 1]; }
        const int bid = (blockIdx.z * gridDim.y + blockIdx.y) * gridDim.x + blockIdx.x;
        blockPartials[bid * 2 + 0] = sw;
        blockPartials[bid * 2 + 1] = sv;
    }
}

extern "C" __global__ __launch_bounds__(256)
void boundary_ce_reduce(const float* __restrict__ parts, int nblocks,
                        float* __restrict__ out)
{
    __shared__ float sws[8], svs[8];
    float sw = 0.0f, sv = 0.0f;
    for (int i = threadIdx.x; i < nblocks; i += 256) {
        sw += parts[2 * i + 0];
        sv += parts[2 * i + 1];
    }
    for (int off = 16; off >= 1; off >>= 1) {
        sw += __shfl_xor(sw, off, 32);
        sv += __shfl_xor(sv, off, 32);
    }
    int lane = threadIdx.x & 31, wid = threadIdx.x >> 5;
    if (lane == 0) { sws[wid] = sw; svs[wid] = sv; }
    __syncthreads();
    if (threadIdx.x == 0) {
        float a = 0.0f, b = 0.0f;
        #pragma unroll
        for (int w = 0; w < 8; ++w) { a += sws[w]; b += svs[w]; }
        const float LN2 = 0.6931471805599453f;  // deferred base-2 -> natural scale
        // reference: nvalid>0 ? sum/max(nvalid,1) : mean(all-zero) == 0
        out[0] = (b > 0.0f) ? ((LN2 * a) / fmaxf(b, 1.0f)) : 0.0f;
    }
}

extern "C" void kernel_launch(void* const* d_in, const int* in_sizes, int n_in,
                              void* d_out, int out_size, void* d_ws, size_t ws_size,
                              hipStream_t stream) {
    const float* logits = (const float*)d_in[0];
    const int*   tgt    = (const int*)d_in[1];
    float* partials = (float*)d_ws;   // 2 floats per block

    const int N = in_sizes[1] / (Hh * Ww);   // 16
    dim3 grid(Ww / TILE, Hh / TILE, N);      // 16 x 16 x 16 = 4096 blocks
    boundary_ce_main<<<grid, 256, 0, stream>>>(logits, tgt, partials);

    const int nblocks = grid.x * grid.y * grid.z;
    boundary_ce_reduce<<<1, 256, 0, stream>>>(partials, nblocks, (float*)d_out);
}